// DynamicNeuralTuringMachine_48241072668793
// MI455X (gfx1250) — compile-verified
//
#include <hip/hip_runtime.h>
#include <math.h>

// D-NTM recurrence for MI455X (gfx1250, wave32).
// 4 persistent workgroups x 256 threads (8 wave32); each WG owns 16 independent
// compute columns and runs the full T=96 recurrence privately. Shared-weight
// GEMMs use V_WMMA_F32_16X16X4_F32 with MULTI-CHAIN interleaving: up to 6
// independent accumulator chains share one LDS B-operand load per K-step, so
// waits amortize 6x and loads pipeline. Per-column memory addressing (cosine
// sim / read / erase-add) is VALU with the 16-column index as the coalesced
// lane axis. contents tile lives in d_ws (L2-resident).

typedef __attribute__((ext_vector_type(2))) float v2f;
typedef __attribute__((ext_vector_type(8))) float v8f;

#define B_  64
#define T_  96
#define F_  64
#define H_  256
#define N_  128
#define C_  96
#define A_  32
#define O_  10
#define M_  128   // C_ + A_
#define BT  16    // compute columns per workgroup
#define NWAVE 8
#define NTHR  256
#define EPSF 1e-8f

__device__ __forceinline__ float sigf(float x) { return 1.0f / (1.0f + expf(-x)); }

#define WMMA_F32(a, b, c) \
  __builtin_amdgcn_wmma_f32_16x16x4_f32(false, (a), false, (b), (short)0, (c), false, false)

// A operand (16x4 f32): lanes 0-15 rows 0-15 hold K={k,k+1}; lanes 16-31 K={k+2,k+3}.
__device__ __forceinline__ const float* arowp(const float* __restrict__ W, int ldw,
                                              int row0, int lane) {
  return W + (size_t)(row0 + (lane & 15)) * ldw + ((lane >> 4) << 1);
}
__device__ __forceinline__ v2f lda(const float* __restrict__ w, int k) {
  v2f a; a.x = w[k]; a.y = w[k + 1]; return a;
}
// B operand (4x16 f32): lanes 0-15 cols hold K rows {k,k+1}; lanes 16-31 {k+2,k+3}.
__device__ __forceinline__ v2f ldb(const float* __restrict__ vc, int k, int kh) {
  v2f b; b.x = vc[(k + kh) * BT]; b.y = vc[(k + kh + 1) * BT]; return b;
}

// Two independent accumulator chains sharing one B stream.
__device__ __forceinline__ void wmma_mac2(v8f& c0, v8f& c1,
    const float* __restrict__ w0, const float* __restrict__ w1,
    const float* __restrict__ V, int K, int lane) {
  const int kh = (lane >> 4) << 1;
  const float* vc = V + (lane & 15);
  for (int k = 0; k < K; k += 4) {
    v2f b = ldb(vc, k, kh);
    c0 = WMMA_F32(lda(w0, k), b, c0);
    c1 = WMMA_F32(lda(w1, k), b, c1);
  }
}

// Six independent accumulator chains sharing one B stream (GRU gate tiles).
__device__ __forceinline__ void wmma_mac6(v8f& c0, v8f& c1, v8f& c2,
                                          v8f& c3, v8f& c4, v8f& c5,
    const float* __restrict__ w0, const float* __restrict__ w1, const float* __restrict__ w2,
    const float* __restrict__ w3, const float* __restrict__ w4, const float* __restrict__ w5,
    const float* __restrict__ V, int K, int lane) {
  const int kh = (lane >> 4) << 1;
  const float* vc = V + (lane & 15);
  for (int k = 0; k < K; k += 4) {
    v2f b = ldb(vc, k, kh);
    c0 = WMMA_F32(lda(w0, k), b, c0);
    c1 = WMMA_F32(lda(w1, k), b, c1);
    c2 = WMMA_F32(lda(w2, k), b, c2);
    c3 = WMMA_F32(lda(w3, k), b, c3);
    c4 = WMMA_F32(lda(w4, k), b, c4);
    c5 = WMMA_F32(lda(w5, k), b, c5);
  }
}

// C/D 16x16 f32 layout: lane col = lane&15; VGPR j -> row j (+8 for lanes 16-31).
__device__ __forceinline__ v8f bias_cd(const float* __restrict__ bias, int row0, int lane) {
  v8f c;
  const int radd = (lane >> 4) << 3;
#pragma unroll
  for (int jj = 0; jj < 8; ++jj) c[jj] = bias[row0 + jj + radd];
  return c;
}
__device__ __forceinline__ void store_cd(float* __restrict__ dst, int row0, v8f c, int lane) {
  const int col  = lane & 15;
  const int radd = (lane >> 4) << 3;
#pragma unroll
  for (int jj = 0; jj < 8; ++jj) dst[(row0 + jj + radd) * BT + col] = c[jj];
}

__global__ __launch_bounds__(NTHR, 1)
void dntm_persistent_kernel(
    const float* __restrict__ batch,       // (B,T,F)
    const float* __restrict__ Wq,          // (M,H)
    const float* __restrict__ bq,          // (M,1)
    const float* __restrict__ us,          // (1,H)
    const float* __restrict__ bs,          // (1,1)
    const float* __restrict__ ul,          // (1,H)
    const float* __restrict__ bl,          // (1,1)
    const float* __restrict__ maddr,       // (N,A)
    const float* __restrict__ Wch,         // (C,H)
    const float* __restrict__ Wci,         // (C,F)
    const float* __restrict__ bc,          // (C,1)
    const float* __restrict__ Weh,         // (C,H)
    const float* __restrict__ Wei,         // (C,F)
    const float* __restrict__ be,          // (C,1)
    const float* __restrict__ Wih,         // (3H,F)
    const float* __restrict__ Whh,         // (3H,H)
    const float* __restrict__ Wmh,         // (3H,M)
    const float* __restrict__ bg,          // (3H,1)
    const float* __restrict__ Wo,          // (O,H)
    const float* __restrict__ bo,          // (O,1)
    float* __restrict__ out,               // hs (T,H,B) then outs (T,O,B)
    float* __restrict__ contents_ws)       // 4 tiles x [n][c][bt] f32
{
  __shared__ float sh_h[H_ * BT];      // hidden state (H x 16)
  __shared__ float sh_x[F_ * BT];      // x_t (F x 16)
  __shared__ float sh_q[M_ * BT];      // query; reused as `reading` after softmax
  __shared__ float sh_cw[N_ * BT];     // logits -> softmax weights (in place)
  __shared__ float sh_ema[N_ * BT];
  __shared__ float sh_cand[C_ * BT];   // reused as output logits in phase 8
  __shared__ float sh_erase[C_ * BT];
  __shared__ float sh_beta[BT], sh_gamma[BT], sh_qn[BT];
  __shared__ float sh_anrm[N_];        // ||addr row||^2 (constant)

  const int tid  = threadIdx.x;
  const int lane = tid & 31;
  const int wave = tid >> 5;
  const int tile = blockIdx.x;         // 0..3
  const int b0   = tile * BT;          // global column base

  float* myct = contents_ws + (size_t)tile * ((size_t)N_ * C_ * BT);

  // ---- init state (deterministic per launch; ws is poisoned) ----
  for (int i = tid; i < H_ * BT; i += NTHR) sh_h[i] = 0.0f;
  for (int i = tid; i < N_ * BT; i += NTHR) sh_ema[i] = 0.0f;
  for (int i = tid; i < N_ * C_ * BT; i += NTHR) myct[i] = 0.0f;
  for (int n = tid; n < N_; n += NTHR) {
    float s = 0.0f;
    for (int a = 0; a < A_; ++a) { float v = maddr[n * A_ + a]; s += v * v; }
    sh_anrm[n] = s;
  }
  __syncthreads();

  const size_t HS_TOT = (size_t)T_ * H_ * B_;

  for (int t = 0; t < T_; ++t) {
    // ---- phase 1: x_t[f][col] = batch[f, t, b0+col] (B==F reshape alias) ----
    for (int i = tid; i < F_ * BT; i += NTHR) {
      int f = i / BT, col = i % BT;
      sh_x[i] = batch[(size_t)f * (T_ * F_) + (size_t)t * F_ + b0 + col];
    }
    __syncthreads();

    // ---- phase 2 (WMMA, 2-chain fused): q ; cand ; erase ; beta/gamma -------
    for (int job = wave; job < 10; job += NWAVE) {
      if (job < 4) {                   // q: fused pair of row tiles (M=128 -> 8)
        int ra = (2 * job) * 16, rb = ra + 16;
        v8f c0 = bias_cd(bq, ra, lane);
        v8f c1 = bias_cd(bq, rb, lane);
        wmma_mac2(c0, c1, arowp(Wq, H_, ra, lane), arowp(Wq, H_, rb, lane),
                  sh_h, H_, lane);
        store_cd(sh_q, ra, c0, lane);
        store_cd(sh_q, rb, c1, lane);
      } else {                         // cand+erase fused per row tile (C=96 -> 6)
        int r0 = (job - 4) * 16;
        v8f cc = bias_cd(bc, r0, lane);
        v8f ce = bias_cd(be, r0, lane);
        wmma_mac2(cc, ce, arowp(Wch, H_, r0, lane), arowp(Weh, H_, r0, lane),
                  sh_h, H_, lane);
        wmma_mac2(cc, ce, arowp(Wci, F_, r0, lane), arowp(Wei, F_, r0, lane),
                  sh_x, F_, lane);
#pragma unroll
        for (int jj = 0; jj < 8; ++jj) {
          cc[jj] = fmaxf(cc[jj], 0.0f);
          ce[jj] = sigf(ce[jj]);
        }
        store_cd(sh_cand, r0, cc, lane);
        store_cd(sh_erase, r0, ce, lane);
      }
    }
    if (wave == 7) {                   // beta (softplus), gamma (sigmoid): 2-way split
      const int colb = lane & 15, half = lane >> 4;
      float sb = 0.0f, sg = 0.0f;
      for (int k = half * (H_ / 2); k < (half + 1) * (H_ / 2); ++k) {
        float hv = sh_h[k * BT + colb];
        sb += us[k] * hv;
        sg += ul[k] * hv;
      }
      sb += __shfl_xor(sb, 16, 32);
      sg += __shfl_xor(sg, 16, 32);
      if (half == 0) {
        sb += bs[0]; sg += bl[0];
        sh_beta[colb]  = (sb > 20.0f) ? sb : log1pf(expf(sb));
        sh_gamma[colb] = sigf(sg);
      }
    }
    __syncthreads();

    // ---- phase 3a: ||q||_2 per column, all 8 waves, 16-lane tree reduce -----
    {
      const int colb = 2 * wave + (lane >> 4);
      const int l16  = lane & 15;
      float s = 0.0f;
#pragma unroll
      for (int i = 0; i < M_ / 16; ++i) {
        float v = sh_q[(l16 + 16 * i) * BT + colb];
        s += v * v;
      }
      s += __shfl_xor(s, 1, 32);
      s += __shfl_xor(s, 2, 32);
      s += __shfl_xor(s, 4, 32);
      s += __shfl_xor(s, 8, 32);
      if (l16 == 0) sh_qn[colb] = sqrtf(s);
    }
    __syncthreads();

    // ---- phase 3b: cosine sim, logits, EMA update (per (n,col), VALU) -------
    {
      const int col = tid & 15;
      for (int n = (tid >> 4); n < N_; n += NTHR / BT) {
        float dot = 0.0f, nrm = sh_anrm[n];
        const float* crow = myct + (size_t)n * (C_ * BT) + col;
        for (int c = 0; c < C_; ++c) {
          float v = crow[c * BT];
          dot += v * sh_q[c * BT + col];
          nrm += v * v;
        }
        for (int a = 0; a < A_; ++a)
          dot += maddr[n * A_ + a] * sh_q[(C_ + a) * BT + col];
        float cosv = dot / (sqrtf(nrm) * sh_qn[col] + EPSF);
        float emao = sh_ema[n * BT + col];
        sh_cw[n * BT + col]  = sh_beta[col] * cosv - sh_gamma[col] * emao;
        sh_ema[n * BT + col] = 0.1f * emao + 0.9f * cosv;
      }
    }
    __syncthreads();

    // ---- phase 4: softmax over N per column; 8 waves x 2 cols, tree reduce --
    {
      const int colb = 2 * wave + (lane >> 4);
      const int l16  = lane & 15;
      float lv[N_ / 16];
      float mx = -3.0e38f;
#pragma unroll
      for (int i = 0; i < N_ / 16; ++i) {
        lv[i] = sh_cw[(l16 + 16 * i) * BT + colb];
        mx = fmaxf(mx, lv[i]);
      }
      mx = fmaxf(mx, __shfl_xor(mx, 1, 32));
      mx = fmaxf(mx, __shfl_xor(mx, 2, 32));
      mx = fmaxf(mx, __shfl_xor(mx, 4, 32));
      mx = fmaxf(mx, __shfl_xor(mx, 8, 32));
      float s = 0.0f;
#pragma unroll
      for (int i = 0; i < N_ / 16; ++i) {
        lv[i] = expf(lv[i] - mx);
        s += lv[i];
      }
      s += __shfl_xor(s, 1, 32);
      s += __shfl_xor(s, 2, 32);
      s += __shfl_xor(s, 4, 32);
      s += __shfl_xor(s, 8, 32);
      float inv = 1.0f / s;
#pragma unroll
      for (int i = 0; i < N_ / 16; ++i)
        sh_cw[(l16 + 16 * i) * BT + colb] = lv[i] * inv;
    }
    __syncthreads();

    // ---- phase 5: reading[m][col] = sum_n mem[col][n][m] * w[n][col] --------
    {
      const int col = tid & 15;
      for (int m = (tid >> 4); m < M_; m += NTHR / BT) {
        float s = 0.0f;
        if (m < C_) {
          const float* cp = myct + (size_t)m * BT + col;
          for (int n = 0; n < N_; ++n)
            s += cp[(size_t)n * (C_ * BT)] * sh_cw[n * BT + col];
        } else {
          int a = m - C_;
          for (int n = 0; n < N_; ++n)
            s += maddr[n * A_ + a] * sh_cw[n * BT + col];
        }
        sh_q[m * BT + col] = s;        // q is dead -> becomes `reading`
      }
    }
    __syncthreads();

    // ---- phase 6: contents erase/add (global, per (n,c,col)) ----------------
    {
      const int col = tid & 15;
      for (int p = (tid >> 4); p < N_ * C_; p += NTHR / BT) {
        int n = p / C_, c = p % C_;
        float wv = sh_cw[n * BT + col];
        size_t idx = ((size_t)n * C_ + c) * BT + col;
        myct[idx] = myct[idx] * (1.0f - wv * sh_erase[c * BT + col])
                  + wv * sh_cand[c * BT + col];
      }
    }

    // ---- phase 7a (WMMA, 6-chain fused): GRU gate tiles in registers --------
    // gates = Wih@x + Wmh@reading + bg ; gh = Whh@h ; rows r/z/n, 2 tiles/wave.
    v8f cr[2], cz[2], cn[2], chn[2];
    {
      const int ra = wave * 16;                  // tile 0 rows
      const int rb = (wave + NWAVE) * 16;        // tile 1 rows
      cr[0] = bias_cd(bg, ra, lane);             cr[1] = bias_cd(bg, rb, lane);
      cz[0] = bias_cd(bg, H_ + ra, lane);        cz[1] = bias_cd(bg, H_ + rb, lane);
      cn[0] = bias_cd(bg, 2 * H_ + ra, lane);    cn[1] = bias_cd(bg, 2 * H_ + rb, lane);
      chn[0] = (v8f){};                          chn[1] = (v8f){};
      // pass over x (K=64): r,z,n for both tiles share one B stream
      wmma_mac6(cr[0], cz[0], cn[0], cr[1], cz[1], cn[1],
                arowp(Wih, F_, ra, lane), arowp(Wih, F_, H_ + ra, lane),
                arowp(Wih, F_, 2 * H_ + ra, lane),
                arowp(Wih, F_, rb, lane), arowp(Wih, F_, H_ + rb, lane),
                arowp(Wih, F_, 2 * H_ + rb, lane),
                sh_x, F_, lane);
      // pass over reading (K=128)
      wmma_mac6(cr[0], cz[0], cn[0], cr[1], cz[1], cn[1],
                arowp(Wmh, M_, ra, lane), arowp(Wmh, M_, H_ + ra, lane),
                arowp(Wmh, M_, 2 * H_ + ra, lane),
                arowp(Wmh, M_, rb, lane), arowp(Wmh, M_, H_ + rb, lane),
                arowp(Wmh, M_, 2 * H_ + rb, lane),
                sh_q, M_, lane);
      // pass over h (K=256): r,z,hn for both tiles
      wmma_mac6(cr[0], cz[0], chn[0], cr[1], cz[1], chn[1],
                arowp(Whh, H_, ra, lane), arowp(Whh, H_, H_ + ra, lane),
                arowp(Whh, H_, 2 * H_ + ra, lane),
                arowp(Whh, H_, rb, lane), arowp(Whh, H_, H_ + rb, lane),
                arowp(Whh, H_, 2 * H_ + rb, lane),
                sh_h, H_, lane);
    }
    __syncthreads();   // all h reads done before any h write

    // ---- phase 7b: GRU elementwise, update h in LDS, emit hs ----------------
    {
      const int col  = lane & 15;
      const int radd = (lane >> 4) << 3;
#pragma unroll
      for (int u = 0; u < 2; ++u) {
        int r0 = (wave + u * NWAVE) * 16;
#pragma unroll
        for (int jj = 0; jj < 8; ++jj) {
          int row = r0 + jj + radd;
          float r  = sigf(cr[u][jj]);
          float z  = sigf(cz[u][jj]);
          float nn = tanhf(cn[u][jj] + r * chn[u][jj]);
          float ho = sh_h[row * BT + col];
          float hn = (1.0f - z) * nn + z * ho;
          sh_h[row * BT + col] = hn;
          out[(size_t)t * (H_ * B_) + (size_t)row * B_ + b0 + col] = hn;
        }
      }
    }
    __syncthreads();

    // ---- phase 8a: output head, K split across 16 lanes + tree reduce -------
    {
      const int colb = 2 * wave + (lane >> 4);
      const int l16  = lane & 15;
      float acc[O_];
#pragma unroll
      for (int o = 0; o < O_; ++o) acc[o] = 0.0f;
      for (int i = 0; i < H_ / 16; ++i) {
        float hv = sh_h[(l16 + 16 * i) * BT + colb];
#pragma unroll
        for (int o = 0; o < O_; ++o) acc[o] += Wo[o * H_ + l16 + 16 * i] * hv;
      }
#pragma unroll
      for (int o = 0; o < O_; ++o) {
        acc[o] += __shfl_xor(acc[o], 1, 32);
        acc[o] += __shfl_xor(acc[o], 2, 32);
        acc[o] += __shfl_xor(acc[o], 4, 32);
        acc[o] += __shfl_xor(acc[o], 8, 32);
        if (l16 == o) sh_cand[o * BT + colb] = acc[o] + bo[o];
      }
    }
    __syncthreads();

    // ---- phase 8b: log_softmax over O=10 per column --------------------------
    if (tid < BT) {
      float mx = -3.0e38f;
      for (int o = 0; o < O_; ++o) mx = fmaxf(mx, sh_cand[o * BT + tid]);
      float se = 0.0f;
      for (int o = 0; o < O_; ++o) se += expf(sh_cand[o * BT + tid] - mx);
      float lse = mx + logf(se);
      for (int o = 0; o < O_; ++o)
        out[HS_TOT + (size_t)t * (O_ * B_) + (size_t)o * B_ + b0 + tid] =
            sh_cand[o * BT + tid] - lse;
    }
    __syncthreads();
  }
}

extern "C" void kernel_launch(void* const* d_in, const int* in_sizes, int n_in,
                              void* d_out, int out_size, void* d_ws, size_t ws_size,
                              hipStream_t stream) {
  (void)in_sizes; (void)n_in; (void)out_size; (void)ws_size;
  const float* p[20];
  for (int i = 0; i < 20; ++i) p[i] = (const float*)d_in[i];
  dntm_persistent_kernel<<<dim3(B_ / BT), dim3(NTHR), 0, stream>>>(
      p[0],  p[1],  p[2],  p[3],  p[4],  p[5],  p[6],  p[7],  p[8],  p[9],
      p[10], p[11], p[12], p[13], p[14], p[15], p[16], p[17], p[18], p[19],
      (float*)d_out, (float*)d_ws);
}